// CartesianLoss_16604343566366
// MI455X (gfx1250) — compile-verified
//
#include <hip/hip_runtime.h>

// ---------------------------------------------------------------------------
// CartesianLoss (permutation-matched assignment loss), MI455X / gfx1250.
// Roofline: 10.5 MB @ 23.3 TB/s ~ 0.45us memory floor; VALU is the real
// limiter, so the hot path uses native v_sin/v_cos/v_sqrt and a subset-DP
// (80 adds + ~50 mins) instead of brute-force 120-permutation search.
// Deterministic two-stage reduction; wave sums use V_WMMA_F32_16X16X4_F32.
// ---------------------------------------------------------------------------

typedef __attribute__((ext_vector_type(2))) float v2f;
typedef __attribute__((ext_vector_type(8))) float v8f;

// Exact f32 sum of one value per lane across the wave (wave32), using WMMA.
// A(16x4) carries the 32 lane values in its first VGPR (other 32 slots = 0),
// B(4x16) = ones, so D[m][n] = rowsum(A)[m]. Sum of all 256 D entries equals
// 16 * sum(A) regardless of the slot<->lane mapping, so this is robust to
// VGPR-layout details. EXEC must be all 1s at the call site.
__device__ __forceinline__ float wave_sum_wmma(float v) {
    v2f a; a[0] = v;    a[1] = 0.0f;   // 32 data slots + 32 zero slots
    v2f b; b[0] = 1.0f; b[1] = 1.0f;   // all-ones 4x16
    v8f c = {0.f, 0.f, 0.f, 0.f, 0.f, 0.f, 0.f, 0.f};
    v8f d = __builtin_amdgcn_wmma_f32_16x16x4_f32(
        /*neg_a=*/false, a, /*neg_b=*/false, b,
        /*c_mod=*/(short)0, c, /*reuse_a=*/false, /*reuse_b=*/false);
    float s = d[0] + d[1] + d[2] + d[3] + d[4] + d[5] + d[6] + d[7];
    s += __shfl_xor(s, 16, 32);
    s += __shfl_xor(s, 8, 32);
    s += __shfl_xor(s, 4, 32);
    s += __shfl_xor(s, 2, 32);
    s += __shfl_xor(s, 1, 32);
    return s * 0.0625f;                // /16: each rowsum replicated 16x in D
}

__global__ void __launch_bounds__(256)
cart_loss_main(const float* __restrict__ pa, const float* __restrict__ ta,
               const float* __restrict__ pd, const float* __restrict__ td,
               float* __restrict__ blk_part, int N) {
    __shared__ float lds[8];
    const int i = blockIdx.x * blockDim.x + threadIdx.x;

    float best = 0.0f;                 // inactive samples contribute 0
    if (i < N) {
        const long base = (long)i * 5;
        float tx[5], ty[5], px[5], py[5];
#pragma unroll
        for (int m = 0; m < 5; ++m) {
            float sa, ca, r;
            // |angle| <= pi/2 -> after *1/(2*pi) the arg is within +/-0.25:
            // prime accuracy range for v_sin_f32 / v_cos_f32.
            __sincosf(ta[base + m], &sa, &ca);
            r = td[base + m];
            tx[m] = ca * r; ty[m] = sa * r;
            __sincosf(pa[base + m], &sa, &ca);
            r = pd[base + m];
            px[m] = ca * r; py[m] = sa * r;
        }
        float D[5][5];
#pragma unroll
        for (int t = 0; t < 5; ++t)
#pragma unroll
            for (int p = 0; p < 5; ++p) {
                const float dx = tx[t] - px[p];
                const float dy = ty[t] - py[p];
                D[t][p] = __builtin_amdgcn_sqrtf(dx * dx + dy * dy); // v_sqrt_f32
            }
        // Held-Karp assignment DP over prediction subsets. Fully unrolled:
        // every index is a compile-time constant, so g[] lives in registers.
        // g[mask] = min cost of assigning targets 0..popcount(mask)-1 to the
        // prediction set `mask`. 80 adds + ~50 mins vs ~450 for brute force.
        float g[32];
#pragma unroll
        for (int mask = 1; mask < 32; ++mask) {
            const int t = __popc(mask) - 1;     // folds to a constant
            float m = 0.0f;
            bool first = true;
#pragma unroll
            for (int p = 0; p < 5; ++p) {
                if (!(mask & (1 << p))) continue;
                const int prev = mask ^ (1 << p);
                const float cand = (prev ? g[prev] : 0.0f) + D[t][p];
                m = first ? cand : fminf(m, cand);
                first = false;
            }
            g[mask] = m;
        }
        best = g[31];                   // optimal assignment cost (sum of 5)
    }
    // EXEC is all 1s here (structured reconvergence, full 256-thread blocks).
    const float ws = wave_sum_wmma(best);
    const int wid  = threadIdx.x >> 5;
    const int lane = threadIdx.x & 31;
    if (lane == 0) lds[wid] = ws;
    __syncthreads();
    if (threadIdx.x == 0) {
        float s = 0.0f;
#pragma unroll
        for (int w = 0; w < 8; ++w) s += lds[w];   // fixed order: deterministic
        blk_part[blockIdx.x] = s;
    }
}

__global__ void __launch_bounds__(32)
cart_loss_finish(const float* __restrict__ part, int nblk,
                 float* __restrict__ out, float inv_den) {
    float s = 0.0f;
    for (int j = threadIdx.x; j < nblk; j += 32) s += part[j];  // fixed order
    const float tot = wave_sum_wmma(s);            // single full wave
    if (threadIdx.x == 0) out[0] = tot * inv_den;  // mean over 5*N distances
}

extern "C" void kernel_launch(void* const* d_in, const int* in_sizes, int n_in,
                              void* d_out, int out_size, void* d_ws, size_t ws_size,
                              hipStream_t stream) {
    const float* pa = (const float*)d_in[0];   // predictions_angle
    const float* ta = (const float*)d_in[1];   // targets_angle
    const float* pd = (const float*)d_in[2];   // predictions_distance
    const float* td = (const float*)d_in[3];   // targets_distance
    const int N = in_sizes[0] / 5;
    const int threads = 256;
    const int blocks = (N + threads - 1) / threads;   // 512 for N=131072

    float* part = (float*)d_ws;                // blocks * 4 bytes of scratch
    cart_loss_main<<<blocks, threads, 0, stream>>>(pa, ta, pd, td, part, N);

    const float inv_den = 1.0f / (5.0f * (float)N);
    cart_loss_finish<<<1, 32, 0, stream>>>(part, blocks, (float*)d_out, inv_den);
}